// DualNeuralNetwork_74603581932111
// MI455X (gfx1250) — compile-verified
//
#include <hip/hip_runtime.h>

typedef float v2f __attribute__((ext_vector_type(2)));
typedef float v8f __attribute__((ext_vector_type(8)));

#define NBINS 28

// Fast sigmoid: v_exp_f32 + v_add_f32 + v_rcp_f32 (avoids the IEEE-division
// div_scale/fma Newton-Raphson chain the compiler emits for `1.0f/x`).
__device__ __forceinline__ float sigmoidf_fast(float x) {
    return __builtin_amdgcn_rcpf(1.0f + __expf(-x));
}

// ---------------------------------------------------------------------------
// Kernel 1: precompute ml-MLP table for dl = 1..40 (exp(mlp(dl) - dl/4)),
// and initialize the 28 output bins with dl2_scores (dl = 1..28).
// One tiny block; runs before the main kernel on the same stream.
// ---------------------------------------------------------------------------
__global__ void ml_table_kernel(const float* __restrict__ W1, const float* __restrict__ b1,
                                const float* __restrict__ W2, const float* __restrict__ b2,
                                const float* __restrict__ W3, const float* __restrict__ b3,
                                float* __restrict__ table, float* __restrict__ out2) {
    int t = threadIdx.x;
    if (t == 0) table[0] = 0.0f;
    if (t >= 1 && t <= 40) {
        float dl = (float)t;
        float h1[16];
        #pragma unroll
        for (int j = 0; j < 16; ++j) h1[j] = sigmoidf_fast(dl * W1[j] + b1[j]);
        float acc = 0.0f;
        #pragma unroll
        for (int j = 0; j < 16; ++j) {
            float s = b2[j];
            #pragma unroll
            for (int i = 0; i < 16; ++i) s += h1[i] * W2[i * 16 + j];
            acc += sigmoidf_fast(s) * W3[j];
        }
        float phi = acc + b3[0];
        float v = __expf(phi - 0.25f * dl);
        table[t] = v;
        if (t <= NBINS) out2[t - 1] = v;   // dl2_scores init (overwritten each call)
    }
}

// ---------------------------------------------------------------------------
// Kernel 2: mh MLP over all rows using V_WMMA_F32_16X16X4_F32.
// Computes transposed activations so C/D layout chains into the next layer's
// B fragments with only half-wave shuffles. 16 samples per wave-tile.
// ---------------------------------------------------------------------------
__global__ void __launch_bounds__(256) mh_wmma_kernel(
    const float* __restrict__ x,
    const float* __restrict__ W1, const float* __restrict__ b1,
    const float* __restrict__ W2, const float* __restrict__ b2,
    const float* __restrict__ W3, const float* __restrict__ b3,
    const int* __restrict__ del_lens, const int* __restrict__ mh_len,
    const float* __restrict__ table,
    float* __restrict__ out, float* __restrict__ out2, int n)
{
    __shared__ float sbins[NBINS];
    __shared__ float stab[41];

    const int tid = threadIdx.x;
    if (tid < NBINS) sbins[tid] = 0.0f;
    if (tid < 41)    stab[tid]  = table[tid];
    __syncthreads();

    const int lane = tid & 31;
    const int c    = lane & 15;     // sample-column within tile / matrix row m
    const int half = lane >> 4;     // 0: lanes 0-15, 1: lanes 16-31

    // --- Preload constant A fragments (W1^T, W2^T) in 16x4 f32 A layout ---
    // A layout: v0 holds K = 4*kk + 2*half, v1 holds K+1; M = lane&15.
    v2f A1[2], A2[4];
    #pragma unroll
    for (int kk = 0; kk < 2; ++kk) {
        int k = 4 * kk + half * 2;
        A1[kk].x = W1[k * 16 + c];          // W1^T[m][k] = W1[k][m]
        A1[kk].y = W1[(k + 1) * 16 + c];
    }
    #pragma unroll
    for (int kk = 0; kk < 4; ++kk) {
        int k = 4 * kk + half * 2;
        A2[kk].x = W2[k * 16 + c];
        A2[kk].y = W2[(k + 1) * 16 + c];
    }
    // Per-lane bias / W3 selects: C/D vgpr r holds row m = r + 8*half.
    float b1s[8], b2s[8], w3s[8];
    #pragma unroll
    for (int r = 0; r < 8; ++r) {
        b1s[r] = b1[r + half * 8];
        b2s[r] = b2[r + half * 8];
        w3s[r] = W3[r + half * 8];
    }
    const float bias3 = b3[0];

    const int ntiles = (n + 15) >> 4;
    const int wavesTotal = (gridDim.x * blockDim.x) >> 5;
    const int waveGlobal = (blockIdx.x * blockDim.x + tid) >> 5;

    for (int tile = waveGlobal; tile < ntiles; tile += wavesTotal) {
        const int s0 = tile << 4;
        const int s  = s0 + c;
        const int sc = (s < n) ? s : (n - 1);   // clamp for tail tile

        // --- Layer 1: B = X^T fragments. Per K-step each lane needs two
        // consecutive features of its sample -> one b64 load each. ---
        const float* xr = x + (size_t)sc * 8 + half * 2;
        v2f B0 = *(const v2f*)(xr);       // features {0,1} / {2,3}
        v2f B1 = *(const v2f*)(xr + 4);   // features {4,5} / {6,7}

        v8f acc = {};
        acc = __builtin_amdgcn_wmma_f32_16x16x4_f32(false, A1[0], false, B0,
                                                    (short)0, acc, false, false);
        acc = __builtin_amdgcn_wmma_f32_16x16x4_f32(false, A1[1], false, B1,
                                                    (short)0, acc, false, false);

        float h[8];
        #pragma unroll
        for (int r = 0; r < 8; ++r) h[r] = sigmoidf_fast(acc[r] + b1s[r]);

        // --- C/D layout -> B fragments for layer 2: half-wave swaps ---
        float sh[8];
        #pragma unroll
        for (int r = 0; r < 8; ++r) sh[r] = __shfl_xor(h[r], 16, 32);

        v8f acc2 = {};
        v2f Bk;
        // kk=0: rows {0,1 | 2,3}
        Bk.x = half ? sh[2] : h[0];  Bk.y = half ? sh[3] : h[1];
        acc2 = __builtin_amdgcn_wmma_f32_16x16x4_f32(false, A2[0], false, Bk,
                                                     (short)0, acc2, false, false);
        // kk=1: rows {4,5 | 6,7}
        Bk.x = half ? sh[6] : h[4];  Bk.y = half ? sh[7] : h[5];
        acc2 = __builtin_amdgcn_wmma_f32_16x16x4_f32(false, A2[1], false, Bk,
                                                     (short)0, acc2, false, false);
        // kk=2: rows {8,9 | 10,11}
        Bk.x = half ? h[2] : sh[0];  Bk.y = half ? h[3] : sh[1];
        acc2 = __builtin_amdgcn_wmma_f32_16x16x4_f32(false, A2[2], false, Bk,
                                                     (short)0, acc2, false, false);
        // kk=3: rows {12,13 | 14,15}
        Bk.x = half ? h[6] : sh[4];  Bk.y = half ? h[7] : sh[5];
        acc2 = __builtin_amdgcn_wmma_f32_16x16x4_f32(false, A2[3], false, Bk,
                                                     (short)0, acc2, false, false);

        // --- Layer 3: 16 -> 1 dot product (each half holds 8 rows) ---
        float p = 0.0f;
        #pragma unroll
        for (int r = 0; r < 8; ++r) p += w3s[r] * sigmoidf_fast(acc2[r] + b2s[r]);
        p += __shfl_xor(p, 16, 32);
        const float phi = p + bias3;

        // --- Epilogue (lanes 0-15 own the 16 samples of this tile) ---
        if (half == 0 && s < n) {
            const int dl = del_lens[s];
            const float score = __expf(phi - 0.25f * (float)dl);
            float y = score;
            if (dl == mh_len[s]) y += stab[dl];       // dl in 1..40
            out[s] = y;
            if (dl <= NBINS) atomicAdd(&sbins[dl - 1], score);
        }
    }

    __syncthreads();
    if (tid < NBINS) atomicAdd(&out2[tid], sbins[tid]);
}

// ---------------------------------------------------------------------------
extern "C" void kernel_launch(void* const* d_in, const int* in_sizes, int n_in,
                              void* d_out, int out_size, void* d_ws, size_t ws_size,
                              hipStream_t stream) {
    const float* x_feat = (const float*)d_in[0];
    const float* mh_W1  = (const float*)d_in[1];
    const float* mh_b1  = (const float*)d_in[2];
    const float* mh_W2  = (const float*)d_in[3];
    const float* mh_b2  = (const float*)d_in[4];
    const float* mh_W3  = (const float*)d_in[5];
    const float* mh_b3  = (const float*)d_in[6];
    const float* ml_W1  = (const float*)d_in[7];
    const float* ml_b1  = (const float*)d_in[8];
    const float* ml_W2  = (const float*)d_in[9];
    const float* ml_b2  = (const float*)d_in[10];
    const float* ml_W3  = (const float*)d_in[11];
    const float* ml_b3  = (const float*)d_in[12];
    const int*   del_lens = (const int*)d_in[13];
    const int*   mh_len   = (const int*)d_in[14];

    const int n = in_sizes[13];            // N rows
    float* out  = (float*)d_out;           // [0, n)  : y_mh_phi
    float* out2 = (float*)d_out + n;       // [n, n+28): y_mh_less_phi
    float* table = (float*)d_ws;           // 41-entry ml LUT

    ml_table_kernel<<<1, 64, 0, stream>>>(ml_W1, ml_b1, ml_W2, ml_b2, ml_W3, ml_b3,
                                          table, out2);

    mh_wmma_kernel<<<1024, 256, 0, stream>>>(x_feat, mh_W1, mh_b1, mh_W2, mh_b2,
                                             mh_W3, mh_b3, del_lens, mh_len,
                                             table, out, out2, n);
}